// MessagePassing_44427141710055
// MI455X (gfx1250) — compile-verified
//
#include <hip/hip_runtime.h>

typedef __attribute__((ext_vector_type(2))) float v2f;
typedef __attribute__((ext_vector_type(4))) float v4f;
typedef __attribute__((ext_vector_type(8))) float v8f;

#define N_NODES 50000
#define N_EDGES 800000
#define D_FEAT  64
#define N_TILES (N_NODES / 16)   // 3125, exact

// ---------------------------------------------------------------------------
// Kernel 1: H = relu(x @ W + b)  (per-node MLP, hoisted out of the edge loop)
// One wave32 computes a 16x64 output tile via V_WMMA_F32_16X16X4_F32.
// Also zeroes the matching 16x64 tile of `out` (atomic accumulation target).
// ---------------------------------------------------------------------------
__global__ __launch_bounds__(256) void node_mlp_relu_kernel(
    const float* __restrict__ x,   // [N_NODES, 64]
    const float* __restrict__ W,   // [64, 64] row-major: W[k][n]
    const float* __restrict__ bias,// [64]
    float* __restrict__ H,         // [N_NODES, 64] workspace
    float* __restrict__ out)       // [N_NODES, 64] zero-initialized here
{
    const int lane = threadIdx.x & 31;
    const int wave = threadIdx.x >> 5;
    const int tile = blockIdx.x * 8 + wave;          // wave-uniform
    if (tile >= N_TILES) return;                     // uniform branch: EXEC stays all-1s

    const int row0 = tile * 16;
    const int half = lane >> 4;                      // 0: K pair {0,1}; 1: K pair {2,3}
    const int l16  = lane & 15;

    // A-fragment base: lane l16 owns row M=l16; halves split the K quad.
    const float* xa = x + (size_t)(row0 + l16) * D_FEAT + 2 * half;

    v8f acc[4] = {v8f{}, v8f{}, v8f{}, v8f{}};

    #pragma unroll
    for (int k = 0; k < D_FEAT; k += 4) {
        // A 16x4 f32 fragment: {x[row][k+2h], x[row][k+2h+1]} -> one b64 load
        v2f a = *(const v2f*)(xa + k);

        // B 4x16 f32 fragments for the 4 N-tiles: lane n=l16 owns column n;
        // VGPR0 = W[k+2h][n], VGPR1 = W[k+2h+1][n]
        const float* wr0 = W + (size_t)(k + 2 * half) * D_FEAT + l16;
        const float* wr1 = wr0 + D_FEAT;

        v2f b0 = { wr0[0],  wr1[0]  };
        v2f b1 = { wr0[16], wr1[16] };
        v2f b2 = { wr0[32], wr1[32] };
        v2f b3 = { wr0[48], wr1[48] };

        acc[0] = __builtin_amdgcn_wmma_f32_16x16x4_f32(false, a, false, b0, (short)0, acc[0], false, false);
        acc[1] = __builtin_amdgcn_wmma_f32_16x16x4_f32(false, a, false, b1, (short)0, acc[1], false, false);
        acc[2] = __builtin_amdgcn_wmma_f32_16x16x4_f32(false, a, false, b2, (short)0, acc[2], false, false);
        acc[3] = __builtin_amdgcn_wmma_f32_16x16x4_f32(false, a, false, b3, (short)0, acc[3], false, false);
    }

    // Bias + ReLU + store. C/D layout: VGPR i -> row (i + 8*half), col l16.
    #pragma unroll
    for (int t = 0; t < 4; ++t) {
        const float bv = bias[t * 16 + l16];
        #pragma unroll
        for (int i = 0; i < 8; ++i) {
            float v = acc[t][i] + bv;
            v = fmaxf(v, 0.0f);
            const int m = i + 8 * half;
            H[(size_t)(row0 + m) * D_FEAT + t * 16 + l16] = v;
        }
    }

    // Zero the matching out tile (1024 floats, 32 lanes -> 8 x v4f per lane)
    float* op = out + (size_t)row0 * D_FEAT;
    #pragma unroll
    for (int i = 0; i < 8; ++i) {
        *(v4f*)(op + i * 128 + lane * 4) = v4f{0.f, 0.f, 0.f, 0.f};
    }
}

// ---------------------------------------------------------------------------
// Kernel 2: out[dst[e]] += H[src[e]]  (edge scatter-add)
// One thread per (edge, 4-float chunk): float4 gather + 4 native f32 atomics.
// H and out both fit in the 192 MB L2, so random access stays on-chip.
// ---------------------------------------------------------------------------
__global__ __launch_bounds__(256) void edge_scatter_kernel(
    const float* __restrict__ H,
    const int* __restrict__ edge_index,   // [2, N_EDGES]: row 0 = src, row 1 = dst
    float* __restrict__ out)
{
    const int tid = blockIdx.x * blockDim.x + threadIdx.x;
    if (tid >= N_EDGES * 16) return;
    const int e = tid >> 4;
    const int c = (tid & 15) << 2;

    const int src = edge_index[e];
    const int dst = edge_index[N_EDGES + e];

    v4f h = *(const v4f*)(H + (size_t)src * D_FEAT + c);
    float* o = out + (size_t)dst * D_FEAT + c;

    unsafeAtomicAdd(o + 0, h.x);
    unsafeAtomicAdd(o + 1, h.y);
    unsafeAtomicAdd(o + 2, h.z);
    unsafeAtomicAdd(o + 3, h.w);
}

extern "C" void kernel_launch(void* const* d_in, const int* in_sizes, int n_in,
                              void* d_out, int out_size, void* d_ws, size_t ws_size,
                              hipStream_t stream) {
    const float* x    = (const float*)d_in[0];   // [50000, 64] f32
    const int*   ei   = (const int*)d_in[1];     // [2, 800000] int
    const float* W    = (const float*)d_in[2];   // [64, 64] f32
    const float* bias = (const float*)d_in[3];   // [64] f32
    float* out = (float*)d_out;                  // [50000, 64] f32
    float* H   = (float*)d_ws;                   // 12.8 MB scratch

    (void)in_sizes; (void)n_in; (void)out_size; (void)ws_size;

    // Phase 1: per-node MLP via WMMA (also zero-fills out)
    const int blocks1 = (N_TILES + 7) / 8;       // 8 waves per 256-thread block
    node_mlp_relu_kernel<<<blocks1, 256, 0, stream>>>(x, W, bias, H, out);

    // Phase 2: edge gather + atomic scatter-add
    const int total2  = N_EDGES * 16;            // 12.8M threads
    const int blocks2 = (total2 + 255) / 256;
    edge_scatter_kernel<<<blocks2, 256, 0, stream>>>(H, ei, out);
}